// HeteroGNN_64527588655555
// MI455X (gfx1250) — compile-verified
//
#include <hip/hip_runtime.h>

#define N_USER 200000
#define N_REPO 100000
#define N_EDGE 1500000
#define DIM 128
#define WSTR 136   // LDS row stride (floats): 128 data + 8 pad -> half-waves hit disjoint banks

typedef __attribute__((ext_vector_type(2))) float v2f;
typedef __attribute__((ext_vector_type(8))) float v8f;
typedef __attribute__((ext_vector_type(4))) unsigned int v4u;
typedef __attribute__((ext_vector_type(8))) int v8i;
typedef __attribute__((ext_vector_type(4))) int v4i;

// ---------------------------------------------------------------------------
// Degree counting: one thread per edge, bump both destination-degree arrays.
// ---------------------------------------------------------------------------
__global__ void count_deg_kernel(const int* __restrict__ uidx,
                                 const int* __restrict__ ridx,
                                 float* __restrict__ cnt_user,
                                 float* __restrict__ cnt_repo,
                                 int nedges) {
    int i = blockIdx.x * blockDim.x + threadIdx.x;
    if (i < nedges) {
        unsafeAtomicAdd(&cnt_user[uidx[i]], 1.0f);   // rev_stars dst = user
        unsafeAtomicAdd(&cnt_repo[ridx[i]], 1.0f);   // stars     dst = repo
    }
}

// ---------------------------------------------------------------------------
// Edge scatter-add: one wave per edge (grid-stride). Each lane moves 4 floats
// (float4 coalesced load -> 4x global_atomic_add_f32 into dst row).
// ---------------------------------------------------------------------------
__global__ void scatter_add_kernel(const float* __restrict__ src,   // [n_src,128]
                                   float* __restrict__ dst_sum,     // [n_dst,128]
                                   const int* __restrict__ sidx,
                                   const int* __restrict__ didx,
                                   int nedges) {
    const int lane   = threadIdx.x & 31;
    const int wave   = (blockIdx.x * blockDim.x + threadIdx.x) >> 5;
    const int nwaves = (gridDim.x * blockDim.x) >> 5;
    for (int e = wave; e < nedges; e += nwaves) {
        const int s = sidx[e];
        const int d = didx[e];
        const float4 v = *(const float4*)(src + (size_t)s * DIM + lane * 4);
        float* dp = dst_sum + (size_t)d * DIM + lane * 4;
        unsafeAtomicAdd(dp + 0, v.x);
        unsafeAtomicAdd(dp + 1, v.y);
        unsafeAtomicAdd(dp + 2, v.z);
        unsafeAtomicAdd(dp + 3, v.w);
    }
}

// ---------------------------------------------------------------------------
// TDM: DMA one 128x128 f32 weight matrix (row-major, contiguous) into LDS,
// inserting 8 DWORDs of padding after every 128 DWORDs (row stride -> 136).
// D# built per ISA 8.3/8.4: group0 = {ctrl, lds_addr, global_addr, type=2},
// group1 = {data_size=4B | pad ctl, dims/tiles/strides}. 2D tensor: groups
// 2/3 (and the extra clang-23 group) zero-filled.
// ---------------------------------------------------------------------------
__device__ __forceinline__ void tdm_load_weight(const float* gsrc, unsigned lds_byte_off) {
    unsigned long long ga = (unsigned long long)gsrc;
    v4u g0;
    g0[0] = 1u;                                          // count=1, user descriptor
    g0[1] = lds_byte_off;                                // lds_addr
    g0[2] = (unsigned)(ga & 0xffffffffu);                // global_addr[31:0]
    g0[3] = (unsigned)((ga >> 32) & 0x01ffffffu)         // global_addr[56:32]
            | 0x80000000u;                               // type=2 ("image")
    v8i g1;
    g1[0] = (2 << 16)                                    // data_size = 4 bytes
          | (1 << 20)                                    // pad_enable
          | (6 << 22)                                    // pad_interval: 128 DWORDs
          | (7 << 25);                                   // pad_amount:   8 DWORDs
    g1[1] = (int)(128u << 16);                           // tensor_dim0[15:0] = 128
    g1[2] = (int)(128u << 16);                           // dim0 MSBs=0 | tensor_dim1 = 128
    g1[3] = (int)(128u << 16);                           // dim1 MSBs=0 | tile_dim0 = 128
    g1[4] = 128;                                         // tile_dim1 = 128, tile_dim2 = 0
    g1[5] = 128;                                         // tensor_dim0_stride = 128
    g1[6] = 0;
    g1[7] = 0;
    v4i z4 = {0, 0, 0, 0};
    v8i z8 = {0, 0, 0, 0, 0, 0, 0, 0};
    // clang-23 / therock-10.0 form: 6 args (g0, g1, g2, g3, g4, cpol)
    __builtin_amdgcn_tensor_load_to_lds(g0, g1, z4, z4, z8, 0);
}

// ---------------------------------------------------------------------------
// Fused SAGE transform via fp32 WMMA, persistent blocks:
//   out = act( (agg_sum/max(cnt,1)) @ Wl + bias + x_dst @ Wr )
// Both weight matrices are TDM-staged into LDS once per block; blocks then
// grid-stride over 16-row M tiles. One wave per 16x16 output tile (8 waves
// = 8 N-tiles of the 128-wide output). Two independent WMMA accumulator
// chains (Wl path seeded with bias, Wr path seeded with 0) for ILP.
// ---------------------------------------------------------------------------
__global__ void __launch_bounds__(256)
sage_wmma_kernel(const float* __restrict__ agg_sum, // [n,128]
                 const float* __restrict__ cnt,     // [n]
                 const float* __restrict__ xdst,    // [n,128]
                 const float* __restrict__ Wl,      // [128,128]
                 const float* __restrict__ bias,    // [128]
                 const float* __restrict__ Wr,      // [128,128]
                 float* __restrict__ out,           // [n,128]
                 int mtiles, int do_relu) {
    __shared__ float lds_wl[DIM * WSTR];
    __shared__ float lds_wr[DIM * WSTR];

    if (threadIdx.x < 32) {            // one wave drives the TDM for the block
        tdm_load_weight(Wl, (unsigned)(unsigned long long)(&lds_wl[0]));
        tdm_load_weight(Wr, (unsigned)(unsigned long long)(&lds_wr[0]));
        __builtin_amdgcn_s_wait_tensorcnt(0);
    }
    __syncthreads();

    const int lane  = threadIdx.x & 31;
    const int nbase = (threadIdx.x >> 5) * 16;        // wave id -> N tile
    const int ncol  = nbase + (lane & 15);            // B/C/D column
    const int kh    = (lane >> 4) * 2;                // K half-select
    const float bv  = bias[ncol];

    for (int mt = blockIdx.x; mt < mtiles; mt += gridDim.x) {
        const int mrow = mt * 16 + (lane & 15);       // A-fragment row
        const float rcp = 1.0f / fmaxf(cnt[mrow], 1.0f);

        const float* mean_row = agg_sum + (size_t)mrow * DIM;
        const float* dst_row  = xdst    + (size_t)mrow * DIM;

        v8f accL, accR;
#pragma unroll
        for (int i = 0; i < 8; ++i) { accL[i] = bv; accR[i] = 0.0f; }

#pragma unroll 4
        for (int k = 0; k < DIM; k += 4) {
            v2f a, b;
            // mean @ Wl (degree reciprocal fused into A fragment)
            a.x = mean_row[k + kh]     * rcp;
            a.y = mean_row[k + kh + 1] * rcp;
            b.x = lds_wl[(k + kh)     * WSTR + ncol];
            b.y = lds_wl[(k + kh + 1) * WSTR + ncol];
            accL = __builtin_amdgcn_wmma_f32_16x16x4_f32(false, a, false, b,
                                                         (short)0, accL, false, false);
            // x_dst @ Wr (independent accumulator chain)
            a.x = dst_row[k + kh];
            a.y = dst_row[k + kh + 1];
            b.x = lds_wr[(k + kh)     * WSTR + ncol];
            b.y = lds_wr[(k + kh + 1) * WSTR + ncol];
            accR = __builtin_amdgcn_wmma_f32_16x16x4_f32(false, a, false, b,
                                                         (short)0, accR, false, false);
        }

        const int mstore = mt * 16 + (lane >> 4) * 8;
#pragma unroll
        for (int i = 0; i < 8; ++i) {
            float r = accL[i] + accR[i];
            if (do_relu) r = fmaxf(r, 0.0f);
            out[(size_t)(mstore + i) * DIM + ncol] = r;
        }
    }
}

// ---------------------------------------------------------------------------
extern "C" void kernel_launch(void* const* d_in, const int* in_sizes, int n_in,
                              void* d_out, int out_size, void* d_ws, size_t ws_size,
                              hipStream_t stream) {
    const float* x_user = (const float*)d_in[0];
    const float* x_repo = (const float*)d_in[1];
    const int*   edges  = (const int*)d_in[2];
    const int*   u_idx  = edges;            // row 0: user indices
    const int*   r_idx  = edges + N_EDGE;   // row 1: repo indices

    const float* Wl1s = (const float*)d_in[3];
    const float* bl1s = (const float*)d_in[4];
    const float* Wr1s = (const float*)d_in[5];
    const float* Wl1r = (const float*)d_in[6];
    const float* bl1r = (const float*)d_in[7];
    const float* Wr1r = (const float*)d_in[8];
    const float* Wl2s = (const float*)d_in[9];
    const float* bl2s = (const float*)d_in[10];
    const float* Wr2s = (const float*)d_in[11];
    const float* Wl2r = (const float*)d_in[12];
    const float* bl2r = (const float*)d_in[13];
    const float* Wr2r = (const float*)d_in[14];

    float* user_out = (float*)d_out;                          // [200k,128]
    float* repo_out = (float*)d_out + (size_t)N_USER * DIM;   // [100k,128]

    // workspace layout (floats)
    float* ws       = (float*)d_ws;
    float* agg_user = ws;                                   size_t o = (size_t)N_USER * DIM;
    float* agg_repo = ws + o;                               o += (size_t)N_REPO * DIM;
    float* user_h   = ws + o;                               o += (size_t)N_USER * DIM;
    float* repo_h   = ws + o;                               o += (size_t)N_REPO * DIM;
    float* cnt_user = ws + o;                               o += N_USER;
    float* cnt_repo = ws + o;                               o += N_REPO;

    const int THREADS = 256;
    const int SC_GRID = 8192;                 // 64k waves grid-striding 1.5M edges
    const int GEMM_GRID = 1024;               // persistent blocks, weights staged once
    const dim3 blk(THREADS);

    // ---- degrees (shared by both layers) + zero layer-1 aggregates ----
    (void)hipMemsetAsync(cnt_user, 0, (size_t)(N_USER + N_REPO) * sizeof(float), stream);
    (void)hipMemsetAsync(agg_user, 0, (size_t)(N_USER + N_REPO) * DIM * sizeof(float), stream);
    count_deg_kernel<<<(N_EDGE + THREADS - 1) / THREADS, blk, 0, stream>>>(
        u_idx, r_idx, cnt_user, cnt_repo, N_EDGE);

    // ---- layer 1 aggregation ----
    scatter_add_kernel<<<SC_GRID, blk, 0, stream>>>(x_user, agg_repo, u_idx, r_idx, N_EDGE);
    scatter_add_kernel<<<SC_GRID, blk, 0, stream>>>(x_repo, agg_user, r_idx, u_idx, N_EDGE);

    // ---- layer 1 transforms (ReLU) ----
    sage_wmma_kernel<<<GEMM_GRID, blk, 0, stream>>>(
        agg_repo, cnt_repo, x_repo, Wl1s, bl1s, Wr1s, repo_h, N_REPO / 16, 1);
    sage_wmma_kernel<<<GEMM_GRID, blk, 0, stream>>>(
        agg_user, cnt_user, x_user, Wl1r, bl1r, Wr1r, user_h, N_USER / 16, 1);

    // ---- layer 2 aggregation ----
    (void)hipMemsetAsync(agg_user, 0, (size_t)(N_USER + N_REPO) * DIM * sizeof(float), stream);
    scatter_add_kernel<<<SC_GRID, blk, 0, stream>>>(user_h, agg_repo, u_idx, r_idx, N_EDGE);
    scatter_add_kernel<<<SC_GRID, blk, 0, stream>>>(repo_h, agg_user, r_idx, u_idx, N_EDGE);

    // ---- layer 2 transforms (no activation), straight into d_out ----
    sage_wmma_kernel<<<GEMM_GRID, blk, 0, stream>>>(
        agg_repo, cnt_repo, repo_h, Wl2s, bl2s, Wr2s, repo_out, N_REPO / 16, 0);
    sage_wmma_kernel<<<GEMM_GRID, blk, 0, stream>>>(
        agg_user, cnt_user, user_h, Wl2r, bl2r, Wr2r, user_out, N_USER / 16, 0);
}